// BallQueryPointGrouping_51127290692137
// MI455X (gfx1250) — compile-verified
//
#include <hip/hip_runtime.h>

#define BB   8
#define NN   8192
#define MM   2048
#define KK   32
#define CIN  64
#define C1   64
#define C2   128
#define RADIUS2 0.01f
#define EPSV    1e-5f
#define NGROUP  (BB*MM)        /* 16384 */
#define NPOS    (BB*MM*KK)     /* 524288 */

typedef __attribute__((ext_vector_type(2))) float v2f;
typedef __attribute__((ext_vector_type(8))) float v8f;

__device__ __forceinline__ v8f wmma4(v2f a, v2f b, v8f c) {
  // D = A(16x4 f32) * B(4x16 f32) + C(16x16 f32)
  return __builtin_amdgcn_wmma_f32_16x16x4_f32(false, a, false, b, (short)0, c, false, false);
}

// ---------------------------------------------------------------------------
// Ball query: one wave32 per query point. First-K-in-order semantics.
// ---------------------------------------------------------------------------
__global__ void bq_kernel(const float* __restrict__ xyz, const float* __restrict__ new_xyz,
                          int* __restrict__ idx_ws, int* __restrict__ valid_ws) {
  const int lane = threadIdx.x & 31;
  const int wave = threadIdx.x >> 5;
  const int g = blockIdx.x * (blockDim.x >> 5) + wave;
  if (g >= NGROUP) return;
  const int b = g / MM;
  const float* q = &new_xyz[(size_t)g * 3];
  const float qx = q[0], qy = q[1], qz = q[2];
  const float qq = qx*qx + qy*qy + qz*qz;
  const float* xb = &xyz[(size_t)b * NN * 3];

  int cnt = 0;
  int myidx = -1;
  for (int n0 = 0; n0 < NN; n0 += 32) {
    const int n = n0 + lane;
    const float px = xb[n*3+0], py = xb[n*3+1], pz = xb[n*3+2];
    const float pp  = px*px + py*py + pz*pz;
    const float dot = qx*px + qy*py + qz*pz;
    const float d2  = qq + pp - 2.0f*dot;          // match reference expanded form
    const unsigned m32 = (unsigned)__ballot(d2 < RADIUS2);
    const int pc = __popc(m32);
    const int myp = lane - cnt;                     // my output slot within this chunk
    if (myp >= 0 && myp < pc) {
      unsigned t = m32;
      for (int i = 0; i < myp; ++i) t &= (t - 1);   // clear myp lowest set bits
      myidx = n0 + (__ffs(t) - 1);
    }
    cnt += pc;
    if (cnt >= KK) break;                            // uniform across wave
  }
  const int has = (cnt > 0);
  const int first = __shfl(myidx, 0, 32);
  const int cc = cnt < KK ? cnt : KK;
  const int v = (lane < cc) ? myidx : (has ? first : 0);
  idx_ws[(size_t)g * KK + lane] = v;
  if (lane == 0) valid_ws[g] = has;
}

// ---------------------------------------------------------------------------
// Fused gather + GEMM1 (+BN1+ReLU + GEMM2) pipeline. One wave32 per group.
// MODE 1: accumulate sum/sumsq of y1 (pre-BN layer 1)
// MODE 2: BN1+ReLU, GEMM2, accumulate sum/sumsq of y2 (pre-BN layer 2)
// MODE 3: BN1+ReLU, GEMM2, BN2+ReLU, validity mask, max over K -> out
// n-tile loops are kept rolled (#pragma unroll 1) with explicit per-tile
// B-strip register preload -> bounded pressure, no scratch spills.
// ---------------------------------------------------------------------------
template<int MODE>
__global__ void pass_kernel(const float* __restrict__ xyz, const float* __restrict__ new_xyz,
                            const float* __restrict__ feats,
                            const float* __restrict__ W1, const float* __restrict__ W2,
                            const int* __restrict__ idx_ws, const int* __restrict__ valid_ws,
                            float* __restrict__ gsum, float* __restrict__ gsq,
                            const float* __restrict__ sc1, const float* __restrict__ sh1,
                            const float* __restrict__ sc2, const float* __restrict__ sh2,
                            float* __restrict__ out) {
  __shared__ float As[2][32 * 68];   // gathered input tile (32 x 67, col 67 zero)
  __shared__ float Hs[2][32 * 68];   // BN1+ReLU activations  (32 x 64)
  __shared__ float sSum[C2];
  __shared__ float sSq[C2];

  const int tid  = threadIdx.x;
  const int lane = tid & 31;
  const int w    = tid >> 5;
  const int half = lane >> 4;
  const int r    = lane & 15;
  constexpr int NC = (MODE == 1) ? C1 : C2;

  if (MODE != 3) {
    for (int t = tid; t < C2; t += blockDim.x) { sSum[t] = 0.f; sSq[t] = 0.f; }
  }
  __syncthreads();

  float* A = As[w];
  float* H = Hs[w];
  const v8f vzero = {0.f,0.f,0.f,0.f,0.f,0.f,0.f,0.f};
  const int nwaves = (gridDim.x * blockDim.x) >> 5;     // 1024

  #pragma unroll 1
  for (int g = blockIdx.x * (blockDim.x >> 5) + w; g < NGROUP; g += nwaves) {
    const int b = g / MM;
    // ---- build A (32 rows x 67 cols, stride 68, col 67 zero) ----
    const int myidx = idx_ws[(size_t)g*KK + lane];
    const float* q = &new_xyz[(size_t)g*3];
    const float qx = q[0], qy = q[1], qz = q[2];
    const float* p = &xyz[((size_t)b*NN + myidx)*3];
    A[lane*68+0] = p[0]-qx; A[lane*68+1] = p[1]-qy; A[lane*68+2] = p[2]-qz;
    A[lane*68+67] = 0.f;
    const float* fb = &feats[(size_t)b*NN*CIN];
    #pragma unroll 4
    for (int rr = 0; rr < KK; ++rr) {
      const int ri = __shfl(myidx, rr, 32);
      const float2* f2 = (const float2*)&fb[(size_t)ri*CIN];
      const float2 vv = f2[lane];
      A[rr*68 + 3 + 2*lane] = vv.x;
      A[rr*68 + 4 + 2*lane] = vv.y;
    }
    __syncthreads();

    // ---- GEMM1: A(32x67) x W1(67x64), rolled n-tile loop ----
    #pragma unroll 1
    for (int nt = 0; nt < 4; ++nt) {
      const int col = nt*16 + r;
      v2f bf[17];                                     // B strip for this column
      #pragma unroll
      for (int ks = 0; ks < 17; ++ks) {
        const int rk = ks*4 + 2*half;
        bf[ks].x = W1[rk*C1 + col];                   // rk <= 66, in range
        bf[ks].y = (rk + 1 < 67) ? W1[(rk+1)*C1 + col] : 0.f;
      }
      v8f acc0 = vzero, acc1 = vzero;
      #pragma unroll
      for (int ks = 0; ks < 17; ++ks) {
        const int rk = ks*4 + 2*half;
        v2f a0, a1;
        a0.x = A[r*68      + rk]; a0.y = A[r*68      + rk + 1];
        a1.x = A[(16+r)*68 + rk]; a1.y = A[(16+r)*68 + rk + 1];
        acc0 = wmma4(a0, bf[ks], acc0);
        acc1 = wmma4(a1, bf[ks], acc1);
      }
      const int c = nt*16 + r;
      if (MODE == 1) {
        float s = 0.f, qs = 0.f;
        #pragma unroll
        for (int vv = 0; vv < 8; ++vv) {
          const float y0 = acc0[vv]; s += y0; qs += y0*y0;
          const float y1 = acc1[vv]; s += y1; qs += y1*y1;
        }
        s  += __shfl_down(s, 16, 32);
        qs += __shfl_down(qs, 16, 32);
        if (lane < 16) { atomicAdd(&sSum[c], s); atomicAdd(&sSq[c], qs); }
      } else {
        const float sc = sc1[c], sh = sh1[c];
        #pragma unroll
        for (int vv = 0; vv < 8; ++vv) {
          float y0 = acc0[vv]*sc + sh; y0 = y0 > 0.f ? y0 : 0.f;
          float y1 = acc1[vv]*sc + sh; y1 = y1 > 0.f ? y1 : 0.f;
          H[(vv + 8*half)*68      + c] = y0;
          H[(16 + vv + 8*half)*68 + c] = y1;
        }
      }
    }

    if (MODE != 1) {
      __syncthreads();
      // ---- GEMM2: H(32x64) x W2(64x128), rolled n-tile loop ----
      const int validg = valid_ws[g];
      #pragma unroll 1
      for (int nt = 0; nt < 8; ++nt) {
        const int col = nt*16 + r;
        v2f bf[16];                                   // B strip for this column
        #pragma unroll
        for (int ks = 0; ks < 16; ++ks) {
          const int rk = ks*4 + 2*half;
          bf[ks].x = W2[rk*C2 + col];
          bf[ks].y = W2[(rk+1)*C2 + col];
        }
        v8f acc0 = vzero, acc1 = vzero;
        #pragma unroll
        for (int ks = 0; ks < 16; ++ks) {
          const int rk = ks*4 + 2*half;
          v2f a0, a1;
          a0.x = H[r*68      + rk]; a0.y = H[r*68      + rk + 1];
          a1.x = H[(16+r)*68 + rk]; a1.y = H[(16+r)*68 + rk + 1];
          acc0 = wmma4(a0, bf[ks], acc0);
          acc1 = wmma4(a1, bf[ks], acc1);
        }
        const int c = nt*16 + r;
        if (MODE == 2) {
          float s = 0.f, qs = 0.f;
          #pragma unroll
          for (int vv = 0; vv < 8; ++vv) {
            const float y0 = acc0[vv]; s += y0; qs += y0*y0;
            const float y1 = acc1[vv]; s += y1; qs += y1*y1;
          }
          s  += __shfl_down(s, 16, 32);
          qs += __shfl_down(qs, 16, 32);
          if (lane < 16) { atomicAdd(&sSum[c], s); atomicAdd(&sSq[c], qs); }
        } else {
          const float sc = sc2[c], sh = sh2[c];
          float cm = -1.f;                            // post-ReLU values >= 0
          #pragma unroll
          for (int vv = 0; vv < 8; ++vv) {
            float y0 = acc0[vv]*sc + sh; y0 = y0 > 0.f ? y0 : 0.f;
            float y1 = acc1[vv]*sc + sh; y1 = y1 > 0.f ? y1 : 0.f;
            const float ym = y0 > y1 ? y0 : y1;
            cm = cm > ym ? cm : ym;
          }
          const float o = __shfl_xor(cm, 16, 32);     // combine row halves
          cm = cm > o ? cm : o;
          if (lane < 16) out[(size_t)g*C2 + c] = validg ? cm : 0.f;
        }
      }
    }
    __syncthreads();
  }

  if (MODE != 3) {
    __syncthreads();
    for (int t = tid; t < NC; t += blockDim.x) {
      atomicAdd(&gsum[t], sSum[t]);
      atomicAdd(&gsq[t],  sSq[t]);
    }
  }
}

// ---------------------------------------------------------------------------
__global__ void zero_kernel(float* __restrict__ p, int n) {
  const int t = blockIdx.x*blockDim.x + threadIdx.x;
  if (t < n) p[t] = 0.f;
}

__global__ void params_kernel(const float* __restrict__ gsum, const float* __restrict__ gsq,
                              const float* __restrict__ gamma, const float* __restrict__ beta,
                              float* __restrict__ scale, float* __restrict__ shift, int nc) {
  const int t = blockIdx.x*blockDim.x + threadIdx.x;
  if (t < nc) {
    const float invP = 1.0f / (float)NPOS;
    const float mean = gsum[t]*invP;
    float var = gsq[t]*invP - mean*mean;
    if (var < 0.f) var = 0.f;
    const float s = gamma[t] * rsqrtf(var + EPSV);
    scale[t] = s;
    shift[t] = beta[t] - mean*s;
  }
}

// ---------------------------------------------------------------------------
extern "C" void kernel_launch(void* const* d_in, const int* in_sizes, int n_in,
                              void* d_out, int out_size, void* d_ws, size_t ws_size,
                              hipStream_t stream) {
  const float* xyz     = (const float*)d_in[0];
  const float* new_xyz = (const float*)d_in[1];
  const float* feats   = (const float*)d_in[2];
  const float* W1      = (const float*)d_in[3];
  const float* g1      = (const float*)d_in[4];
  const float* b1      = (const float*)d_in[5];
  const float* W2      = (const float*)d_in[6];
  const float* g2      = (const float*)d_in[7];
  const float* b2      = (const float*)d_in[8];
  float* out = (float*)d_out;

  char* ws = (char*)d_ws;
  int*   idx_ws   = (int*)ws;                                        // NGROUP*KK ints
  int*   valid_ws = (int*)(ws + (size_t)NGROUP*KK*sizeof(int));      // NGROUP ints
  float* statf    = (float*)(ws + (size_t)NGROUP*KK*sizeof(int) + (size_t)NGROUP*sizeof(int));
  float* sum1 = statf +   0;  float* sq1 = statf +  64;
  float* sum2 = statf + 128;  float* sq2 = statf + 256;
  float* sc1  = statf + 384;  float* sh1 = statf + 448;
  float* sc2  = statf + 512;  float* sh2 = statf + 640;

  zero_kernel<<<1, 512, 0, stream>>>(statf, 384);
  bq_kernel<<<NGROUP/8, 256, 0, stream>>>(xyz, new_xyz, idx_ws, valid_ws);
  pass_kernel<1><<<512, 64, 0, stream>>>(xyz, new_xyz, feats, W1, W2, idx_ws, valid_ws,
                                         sum1, sq1, sc1, sh1, sc2, sh2, out);
  params_kernel<<<1, 64, 0, stream>>>(sum1, sq1, g1, b1, sc1, sh1, 64);
  pass_kernel<2><<<512, 64, 0, stream>>>(xyz, new_xyz, feats, W1, W2, idx_ws, valid_ws,
                                         sum2, sq2, sc1, sh1, sc2, sh2, out);
  params_kernel<<<1, 128, 0, stream>>>(sum2, sq2, g2, b2, sc2, sh2, 128);
  pass_kernel<3><<<512, 64, 0, stream>>>(xyz, new_xyz, feats, W1, W2, idx_ws, valid_ws,
                                         sum2, sq2, sc1, sh1, sc2, sh2, out);
}